// GATClassifier_52175262712210
// MI455X (gfx1250) — compile-verified
//
#include <hip/hip_runtime.h>
#include <hip/hip_bf16.h>
#include <math.h>

#define NN 50000
#define EE 800000
#define GG 512
#define INF_DIM 544
#define HIDD 64
#define HEADS 4
#define EPSV 1e-5f

typedef float v2f __attribute__((ext_vector_type(2)));
typedef float v8f __attribute__((ext_vector_type(8)));

// ---- workspace layout (float offsets) ----
constexpr long long OFF_BN1  = 0;                       // 512 sum + 512 sumsq
constexpr long long OFF_WCP  = 1024;                    // 16*512 folded Wc
constexpr long long OFF_BCP  = 9216;                    // 16
constexpr long long OFF_BN2  = 9232;                    // 48 sum + 48 sumsq
constexpr long long OFF_WPP  = 9328;                    // 64*48 folded Wp
constexpr long long OFF_BPP  = 12400;                   // 64
constexpr long long OFF_BN3  = 12464;                   // 256 sum + 256 sumsq
constexpr long long OFF_SC3  = 12976;                   // 256
constexpr long long OFF_SH3  = 13232;                   // 256
constexpr long long OFF_CNT  = 13488;                   // 512
constexpr long long OFF_ZPRE = 14080;                   // N*48
constexpr long long OFF_H    = OFF_ZPRE + (long long)NN*48;    // N*256
constexpr long long OFF_OUT  = OFF_H    + (long long)NN*256;   // N*256 accum
constexpr long long OFF_ASRC = OFF_OUT  + (long long)NN*256;   // N*4
constexpr long long OFF_ADST = OFF_ASRC + (long long)NN*4;     // N*4
constexpr long long OFF_M    = OFF_ADST + (long long)NN*4;     // N*4 (init -inf)
constexpr long long OFF_S    = OFF_M    + (long long)NN*4;     // N*4
constexpr long long OFF_E    = OFF_S    + (long long)NN*4;     // E*4
constexpr long long OFF_POOL = OFF_E    + (long long)EE*4;     // G*256
constexpr long long WS_TOTAL = OFF_POOL + (long long)GG*256;

static __device__ __forceinline__ v8f wmma4(v2f a, v2f b, v8f c) {
  return __builtin_amdgcn_wmma_f32_16x16x4_f32(false, a, false, b, (short)0, c, false, false);
}

static __device__ __forceinline__ void atomicMaxF(float* addr, float val) {
  int* ia = (int*)addr;
  int old = *ia;
  while (__int_as_float(old) < val) {
    int prev = atomicCAS(ia, old, __float_as_int(val));
    if (prev == old) break;
    old = prev;
  }
}

// ---- init: zero workspace, m = -inf ----
__global__ void k_init(float* ws) {
  long long i = (long long)blockIdx.x * blockDim.x + threadIdx.x;
  if (i >= WS_TOTAL) return;
  ws[i] = (i >= OFF_M && i < OFF_M + (long long)NN*4) ? -INFINITY : 0.0f;
}

// ---- BN1 column stats over x[:, :512] ----
__global__ __launch_bounds__(256) void k_bn1red(const float* __restrict__ x, float* __restrict__ ws) {
  int col = blockIdx.x * 256 + threadIdx.x;            // gridDim.x = 2
  long long r0 = (long long)blockIdx.y * 250;          // gridDim.y = 200
  float s = 0.f, q = 0.f;
  for (long long r = r0; r < r0 + 250; ++r) {
    float v = x[r * INF_DIM + col];
    s += v; q += v * v;
  }
  atomicAdd(&ws[OFF_BN1 + col], s);
  atomicAdd(&ws[OFF_BN1 + 512 + col], q);
}

// ---- fold BN1 into Wc:  Wc' = Wc*scale,  bc' = bc + Wc.(b - mu*scale) ----
__global__ __launch_bounds__(256) void k_foldWc(const float* __restrict__ Wc, const float* __restrict__ bc,
                                                const float* __restrict__ g1, const float* __restrict__ b1v,
                                                float* __restrict__ ws) {
  __shared__ float red[256];
  int c = blockIdx.x, t = threadIdx.x;
  float acc = 0.f;
  for (int j = t; j < 512; j += 256) {
    float mu = ws[OFF_BN1 + j] * (1.0f / NN);
    float var = ws[OFF_BN1 + 512 + j] * (1.0f / NN) - mu * mu;
    float scale = g1[j] * rsqrtf(var + EPSV);
    float w = Wc[c * 512 + j];
    ws[OFF_WCP + c * 512 + j] = w * scale;
    acc += (b1v[j] - mu * scale) * w;
  }
  red[t] = acc; __syncthreads();
  for (int o = 128; o > 0; o >>= 1) { if (t < o) red[t] += red[t + o]; __syncthreads(); }
  if (t == 0) ws[OFF_BCP + c] = bc[c] + red[0];
}

// ---- layernorm of x[:, 512:544] -> zpre[:, 16:48] (wave per node) ----
__global__ __launch_bounds__(256) void k_lnf(const float* __restrict__ x, const float* __restrict__ lng,
                                             const float* __restrict__ lnb, float* __restrict__ ws) {
  int lane = threadIdx.x & 31;
  long long n = (long long)blockIdx.x * 8 + (threadIdx.x >> 5);
  float v = x[n * INF_DIM + 512 + lane];
  float s = v, q = v * v;
  for (int o = 16; o > 0; o >>= 1) { s += __shfl_xor(s, o, 32); q += __shfl_xor(q, o, 32); }
  float mean = s * (1.0f / 32.0f);
  float var = q * (1.0f / 32.0f) - mean * mean;
  ws[OFF_ZPRE + n * 48 + 16 + lane] = (v - mean) * rsqrtf(var + EPSV) * lng[lane] + lnb[lane];
}

// ---- x(16x512) @ Wc'^T -> zpre[:, :16]  via f32 WMMA 16x16x4, one wave per 16-node tile ----
__global__ __launch_bounds__(32) void k_xcnn(const float* __restrict__ x, float* __restrict__ ws) {
  const int lane = threadIdx.x & 31;
  const int half = lane >> 4;
  const int mn = lane & 15;
  const long long base = (long long)blockIdx.x * 16;
  const float* __restrict__ Wcp = ws + OFF_WCP;
  v8f c;
  float bias = ws[OFF_BCP + mn];
#pragma unroll
  for (int r = 0; r < 8; ++r) c[r] = bias;
  const float* arow = x + (base + mn) * INF_DIM;       // A row m=mn
  const float* bcol = Wcp + (long long)mn * 512;       // B col n=mn (B[k][n] = Wc'[n][k])
  for (int k = 0; k < 512; k += 4) {
    v2f a, b;
    a[0] = arow[k + 2 * half]; a[1] = arow[k + 2 * half + 1];
    b[0] = bcol[k + 2 * half]; b[1] = bcol[k + 2 * half + 1];
    c = wmma4(a, b, c);
  }
#pragma unroll
  for (int r = 0; r < 8; ++r)
    ws[OFF_ZPRE + (base + r + 8 * half) * 48 + mn] = c[r];
}

// ---- BN2 column stats over zpre (48 cols) ----
__global__ __launch_bounds__(64) void k_bn2red(float* __restrict__ ws) {
  int col = threadIdx.x;
  if (col >= 48) return;
  long long r0 = (long long)blockIdx.y * 500;          // gridDim.y = 100
  float s = 0.f, q = 0.f;
  for (long long r = r0; r < r0 + 500; ++r) {
    float v = ws[OFF_ZPRE + r * 48 + col];
    s += v; q += v * v;
  }
  atomicAdd(&ws[OFF_BN2 + col], s);
  atomicAdd(&ws[OFF_BN2 + 48 + col], q);
}

// ---- fold BN2 into Wp ----
__global__ __launch_bounds__(64) void k_foldWp(const float* __restrict__ Wp, const float* __restrict__ bp,
                                               const float* __restrict__ g2, const float* __restrict__ b2v,
                                               float* __restrict__ ws) {
  __shared__ float red[64];
  int o = blockIdx.x, t = threadIdx.x;
  float acc = 0.f;
  if (t < 48) {
    float mu = ws[OFF_BN2 + t] * (1.0f / NN);
    float var = ws[OFF_BN2 + 48 + t] * (1.0f / NN) - mu * mu;
    float scale = g2[t] * rsqrtf(var + EPSV);
    float w = Wp[o * 48 + t];
    ws[OFF_WPP + o * 48 + t] = w * scale;
    acc = (b2v[t] - mu * scale) * w;
  }
  red[t] = acc; __syncthreads();
  for (int s = 32; s > 0; s >>= 1) { if (t < s) red[t] += red[t + s]; __syncthreads(); }
  if (t == 0) ws[OFF_BPP + o] = bp[o] + red[0];
}

// ---- zr = relu(zpre @ Wp'^T + bp'), h = zr @ Wg^T : all WMMA, one wave per 16-node tile ----
__global__ __launch_bounds__(32) void k_hfeat(float* __restrict__ ws, const float* __restrict__ Wg) {
  __shared__ float zr_s[16 * 64];
  const int lane = threadIdx.x & 31;
  const int half = lane >> 4;
  const int mn = lane & 15;
  const long long base = (long long)blockIdx.x * 16;
  const float* arow = ws + OFF_ZPRE + (base + mn) * 48;
  // stage 1: 16x48 @ 48x64
#pragma unroll
  for (int nt = 0; nt < 4; ++nt) {
    v8f c;
    float bias = ws[OFF_BPP + nt * 16 + mn];
#pragma unroll
    for (int r = 0; r < 8; ++r) c[r] = bias;
    const float* bcol = ws + OFF_WPP + (long long)(nt * 16 + mn) * 48;
    for (int k = 0; k < 48; k += 4) {
      v2f a, b;
      a[0] = arow[k + 2 * half]; a[1] = arow[k + 2 * half + 1];
      b[0] = bcol[k + 2 * half]; b[1] = bcol[k + 2 * half + 1];
      c = wmma4(a, b, c);
    }
#pragma unroll
    for (int r = 0; r < 8; ++r)
      zr_s[(r + 8 * half) * 64 + nt * 16 + mn] = fmaxf(c[r], 0.0f);
  }
  __syncthreads();
  // stage 2: 16x64 @ 64x256
  for (int nt = 0; nt < 16; ++nt) {
    v8f c = (v8f){0.f,0.f,0.f,0.f,0.f,0.f,0.f,0.f};
    const float* bcol = Wg + (long long)(nt * 16 + mn) * 64;
    for (int k = 0; k < 64; k += 4) {
      v2f a, b;
      a[0] = zr_s[mn * 64 + k + 2 * half]; a[1] = zr_s[mn * 64 + k + 2 * half + 1];
      b[0] = bcol[k + 2 * half];           b[1] = bcol[k + 2 * half + 1];
      c = wmma4(a, b, c);
    }
#pragma unroll
    for (int r = 0; r < 8; ++r)
      ws[OFF_H + (base + r + 8 * half) * 256 + nt * 16 + mn] = c[r];
  }
}

// ---- attention logits a_src/a_dst (wave per node) ----
__global__ __launch_bounds__(256) void k_att(float* __restrict__ ws, const float* __restrict__ as,
                                             const float* __restrict__ ad) {
  int lane = threadIdx.x & 31;
  long long n = (long long)blockIdx.x * 8 + (threadIdx.x >> 5);
  const float* hr = ws + OFF_H + n * 256;
#pragma unroll
  for (int hh = 0; hh < HEADS; ++hh) {
    int c0 = hh * 64 + lane, c1 = hh * 64 + 32 + lane;
    float vs = hr[c0] * as[c0] + hr[c1] * as[c1];
    float vd = hr[c0] * ad[c0] + hr[c1] * ad[c1];
    for (int o = 16; o > 0; o >>= 1) { vs += __shfl_xor(vs, o, 32); vd += __shfl_xor(vd, o, 32); }
    if (lane == 0) {
      ws[OFF_ASRC + n * 4 + hh] = vs;
      ws[OFF_ADST + n * 4 + hh] = vd;
    }
  }
}

// ---- edge pass 1: e = leaky_relu(a_src[src]+a_dst[dst]); segment max into m ----
__global__ __launch_bounds__(256) void k_edge1(const int* __restrict__ ei, float* __restrict__ ws) {
  long long t = (long long)blockIdx.x * 256 + threadIdx.x;   // E*4 threads
  long long e = t >> 2; int hh = (int)(t & 3);
  long long s = ei[e], d = ei[EE + e];
  float v = ws[OFF_ASRC + s * 4 + hh] + ws[OFF_ADST + d * 4 + hh];
  v = (v > 0.f) ? v : 0.2f * v;
  ws[OFF_E + t] = v;
  atomicMaxF(&ws[OFF_M + d * 4 + hh], v);
}

// ---- edge pass 2: ex = exp(e - m[dst]); segment sum into s ----
__global__ __launch_bounds__(256) void k_edge2(const int* __restrict__ ei, float* __restrict__ ws) {
  long long t = (long long)blockIdx.x * 256 + threadIdx.x;
  long long e = t >> 2; int hh = (int)(t & 3);
  long long d = ei[EE + e];
  float mm = ws[OFF_M + d * 4 + hh];
  if (!(mm >= -3.0e38f)) mm = 0.0f;                    // -inf (no edges) -> 0, per reference
  float ex = expf(ws[OFF_E + t] - mm);
  ws[OFF_E + t] = ex;
  atomicAdd(&ws[OFF_S + d * 4 + hh], ex);
}

// ---- edge pass 3: out[dst] += alpha * h[src]  (wave per edge, 8 chans per lane) ----
__global__ __launch_bounds__(256) void k_edge3(const int* __restrict__ ei, float* __restrict__ ws) {
  int lane = threadIdx.x & 31;
  long long e = (long long)blockIdx.x * 8 + (threadIdx.x >> 5);
  long long s = ei[e], d = ei[EE + e];
  const float* hs = ws + OFF_H + s * 256;
  float* od = ws + OFF_OUT + d * 256;
#pragma unroll
  for (int it = 0; it < 8; ++it) {
    int c = lane + it * 32;
    int hh = c >> 6;
    float ssum = ws[OFF_S + d * 4 + hh];
    float alpha = ws[OFF_E + e * 4 + hh] / fmaxf(ssum, 1e-16f);
    atomicAdd(&od[c], alpha * hs[c]);
  }
}

// ---- BN3 column stats over (out + bg) ----
__global__ __launch_bounds__(256) void k_bn3red(float* __restrict__ ws, const float* __restrict__ bg) {
  int col = threadIdx.x;
  long long r0 = (long long)blockIdx.y * 250;          // gridDim.y = 200
  float bgv = bg[col];
  float s = 0.f, q = 0.f;
  for (long long r = r0; r < r0 + 250; ++r) {
    float v = ws[OFF_OUT + r * 256 + col] + bgv;
    s += v; q += v * v;
  }
  atomicAdd(&ws[OFF_BN3 + col], s);
  atomicAdd(&ws[OFF_BN3 + 256 + col], q);
}

// ---- finalize BN3 scale/shift (bg folded in) ----
__global__ void k_sc3(const float* __restrict__ g3, const float* __restrict__ b3,
                      const float* __restrict__ bg, float* __restrict__ ws) {
  int c = threadIdx.x;
  float mu = ws[OFF_BN3 + c] * (1.0f / NN);
  float var = ws[OFF_BN3 + 256 + c] * (1.0f / NN) - mu * mu;
  float scale = g3[c] * rsqrtf(var + EPSV);
  ws[OFF_SC3 + c] = scale;
  ws[OFF_SH3 + c] = (bg[c] - mu) * scale + b3[c];
}

// ---- BN3 + ELU + mean-pool accumulate (wave per node) ----
__global__ __launch_bounds__(256) void k_pool(float* __restrict__ ws, const int* __restrict__ batch) {
  int lane = threadIdx.x & 31;
  long long n = (long long)blockIdx.x * 8 + (threadIdx.x >> 5);
  long long b = batch[n];
#pragma unroll
  for (int it = 0; it < 8; ++it) {
    int c = lane + it * 32;
    float v = ws[OFF_OUT + n * 256 + c] * ws[OFF_SC3 + c] + ws[OFF_SH3 + c];
    float o = (v > 0.f) ? v : (expf(v) - 1.0f);
    atomicAdd(&ws[OFF_POOL + b * 256 + c], o);
  }
  if (lane == 0) atomicAdd(&ws[OFF_CNT + b], 1.0f);
}

// ---- per-graph MLP: relu(pooled/cnt @ W1^T + b1) @ W2^T + b2 ----
__global__ __launch_bounds__(64) void k_mlp(const float* __restrict__ ws, const float* __restrict__ W1,
                                            const float* __restrict__ b1, const float* __restrict__ W2,
                                            const float* __restrict__ b2, float* __restrict__ out) {
  __shared__ float hm[64];
  int g = blockIdx.x, t = threadIdx.x;
  float inv = 1.0f / fmaxf(ws[OFF_CNT + g], 1.0f);
  float acc = b1[t];
  for (int c = 0; c < 256; ++c)
    acc += ws[OFF_POOL + (long long)g * 256 + c] * inv * W1[t * 256 + c];
  hm[t] = fmaxf(acc, 0.0f);
  __syncthreads();
  if (t < 2) {
    float o = b2[t];
    for (int k = 0; k < 64; ++k) o += hm[k] * W2[t * 64 + k];
    out[g * 2 + t] = o;
  }
}

extern "C" void kernel_launch(void* const* d_in, const int* in_sizes, int n_in,
                              void* d_out, int out_size, void* d_ws, size_t ws_size,
                              hipStream_t stream) {
  const float* x       = (const float*)d_in[0];
  const int*   ei      = (const int*)d_in[1];
  const int*   batch   = (const int*)d_in[2];
  const float* ln_g    = (const float*)d_in[3];
  const float* ln_b    = (const float*)d_in[4];
  const float* bn1_g   = (const float*)d_in[5];
  const float* bn1_b   = (const float*)d_in[6];
  const float* Wc      = (const float*)d_in[7];
  const float* bc      = (const float*)d_in[8];
  const float* bn2_g   = (const float*)d_in[9];
  const float* bn2_b   = (const float*)d_in[10];
  const float* Wp      = (const float*)d_in[11];
  const float* bp      = (const float*)d_in[12];
  const float* Wg      = (const float*)d_in[13];
  const float* att_src = (const float*)d_in[14];
  const float* att_dst = (const float*)d_in[15];
  const float* bg      = (const float*)d_in[16];
  const float* bn3_g   = (const float*)d_in[17];
  const float* bn3_b   = (const float*)d_in[18];
  const float* W1      = (const float*)d_in[19];
  const float* b1      = (const float*)d_in[20];
  const float* W2      = (const float*)d_in[21];
  const float* b2      = (const float*)d_in[22];
  float* ws  = (float*)d_ws;
  float* out = (float*)d_out;

  int initBlocks = (int)((WS_TOTAL + 255) / 256);
  k_init<<<initBlocks, 256, 0, stream>>>(ws);
  k_bn1red<<<dim3(2, 200), 256, 0, stream>>>(x, ws);
  k_foldWc<<<16, 256, 0, stream>>>(Wc, bc, bn1_g, bn1_b, ws);
  k_lnf<<<NN / 8, 256, 0, stream>>>(x, ln_g, ln_b, ws);
  k_xcnn<<<NN / 16, 32, 0, stream>>>(x, ws);
  k_bn2red<<<dim3(1, 100), 64, 0, stream>>>(ws);
  k_foldWp<<<64, 64, 0, stream>>>(Wp, bp, bn2_g, bn2_b, ws);
  k_hfeat<<<NN / 16, 32, 0, stream>>>(ws, Wg);
  k_att<<<NN / 8, 256, 0, stream>>>(ws, att_src, att_dst);
  k_edge1<<<(EE * 4) / 256, 256, 0, stream>>>(ei, ws);
  k_edge2<<<(EE * 4) / 256, 256, 0, stream>>>(ei, ws);
  k_edge3<<<EE / 8, 256, 0, stream>>>(ei, ws);
  k_bn3red<<<dim3(1, 200), 256, 0, stream>>>(ws, bg);
  k_sc3<<<1, 256, 0, stream>>>(bn3_g, bn3_b, bg, ws);
  k_pool<<<NN / 8, 256, 0, stream>>>(ws, batch);
  k_mlp<<<GG, 64, 0, stream>>>(ws, W1, b1, W2, b2, out);
}